// QuanvolutionHybrid_65481071397322
// MI455X (gfx1250) — compile-verified
//
#include <hip/hip_runtime.h>
#include <math.h>

typedef float v2f __attribute__((ext_vector_type(2)));
typedef float v8f __attribute__((ext_vector_type(8)));

#define NPATCH 196

// ---------------------------------------------------------------------------
// Kernel 1: build the 4 real symmetric 16x16 matrices M_w = Re(U^H D_w U)
// from the 4 scalar circuit params. One block, 256 threads; 16 threads each
// simulate one column of U (basis state through the fixed gate sequence).
// ---------------------------------------------------------------------------
__global__ void build_M_kernel(const float* __restrict__ qp, float* __restrict__ Mg) {
    __shared__ float Ure[16][16]; // [m][col]
    __shared__ float Uim[16][16];
    const int t = threadIdx.x;

    if (t < 16) {
        float re[16], im[16];
        #pragma unroll
        for (int m = 0; m < 16; ++m) { re[m] = 0.f; im[m] = 0.f; }
        re[t] = 1.f;

        const float q0 = qp[0], q1 = qp[1], q2 = qp[2], q3 = qp[3];

        auto ry = [&](float th, int wire) {
            float s, c; sincosf(0.5f * th, &s, &c);
            const int mask = 1 << (3 - wire);
            #pragma unroll
            for (int m = 0; m < 16; ++m) {
                if (m & mask) continue;
                const int m1 = m | mask;
                float r0 = re[m], i0 = im[m], r1 = re[m1], i1 = im[m1];
                re[m]  = c * r0 - s * r1;  im[m]  = c * i0 - s * i1;
                re[m1] = s * r0 + c * r1;  im[m1] = s * i0 + c * i1;
            }
        };
        auto rx = [&](float th, int wire) {
            float s, c; sincosf(0.5f * th, &s, &c);
            const int mask = 1 << (3 - wire);
            #pragma unroll
            for (int m = 0; m < 16; ++m) {
                if (m & mask) continue;
                const int m1 = m | mask;
                float r0 = re[m], i0 = im[m], r1 = re[m1], i1 = im[m1];
                // a0' = c*a0 - i*s*a1 ; a1' = -i*s*a0 + c*a1
                re[m]  = c * r0 + s * i1;   im[m]  = c * i0 - s * r1;
                re[m1] = s * i0 + c * r1;   im[m1] = -s * r0 + c * i1;
            }
        };
        auto cnot = [&](int cw, int tw) {
            const int cm = 1 << (3 - cw), tm = 1 << (3 - tw);
            #pragma unroll
            for (int m = 0; m < 16; ++m) {
                if ((m & cm) && !(m & tm)) {
                    const int m1 = m | tm;
                    float r = re[m]; re[m] = re[m1]; re[m1] = r;
                    float i = im[m]; im[m] = im[m1]; im[m1] = i;
                }
            }
        };

        ry(q0, 0); rx(q1, 1); cnot(0, 1);
        ry(q2, 2); cnot(1, 2);
        rx(q3, 3); cnot(2, 3); cnot(3, 0);

        #pragma unroll
        for (int m = 0; m < 16; ++m) { Ure[m][t] = re[m]; Uim[m][t] = im[m]; }
    }
    __syncthreads();

    const int j = t >> 4, k = t & 15;
    float a0 = 0.f, a1 = 0.f, a2 = 0.f, a3 = 0.f;
    #pragma unroll
    for (int m = 0; m < 16; ++m) {
        const float pr = Ure[m][j] * Ure[m][k] + Uim[m][j] * Uim[m][k];
        a0 += (m & 8) ? -pr : pr;   // wire 0 <-> bit 3 (MSB)
        a1 += (m & 4) ? -pr : pr;
        a2 += (m & 2) ? -pr : pr;
        a3 += (m & 1) ? -pr : pr;
    }
    Mg[0 * 256 + j * 16 + k] = a0;
    Mg[1 * 256 + j * 16 + k] = a1;
    Mg[2 * 256 + j * 16 + k] = a2;
    Mg[3 * 256 + j * 16 + k] = a3;
}

// ---------------------------------------------------------------------------
// Kernel 2: per (batch, patch) quadratic forms  meas_w = v^T M_w v  where v is
// the rank-1 real encoder state. One thread per (b,p); M_w broadcast from LDS.
// ---------------------------------------------------------------------------
__global__ void __launch_bounds__(256)
quanv_kernel(const float* __restrict__ x, const float* __restrict__ Mg,
             float* __restrict__ feat, int total) {
    __shared__ float Ms[4][16][16];
    const int t = threadIdx.x;
    #pragma unroll
    for (int i = 0; i < 4; ++i) ((float*)Ms)[t + 256 * i] = Mg[t + 256 * i];
    __syncthreads();

    const int gid = blockIdx.x * 256 + t;
    if (gid >= total) return;

    const int b = gid / NPATCH, p = gid - b * NPATCH;
    const int r = p / 14, c = p - r * 14;
    const float* xb = x + (size_t)b * 784 + r * 56 + c * 2;
    const float px0 = xb[0], px1 = xb[1], px2 = xb[28], px3 = xb[29];

    float g[4][2]; // g[w][0]=cos(x/2), g[w][1]=sin(x/2)
    sincosf(0.5f * px0, &g[0][1], &g[0][0]);
    sincosf(0.5f * px1, &g[1][1], &g[1][0]);
    sincosf(0.5f * px2, &g[2][1], &g[2][0]);
    sincosf(0.5f * px3, &g[3][1], &g[3][0]);

    float v[16];
    #pragma unroll
    for (int k = 0; k < 16; ++k)
        v[k] = g[0][(k >> 3) & 1] * g[1][(k >> 2) & 1] * g[2][(k >> 1) & 1] * g[3][k & 1];

    float* fo = feat + (size_t)b * 784 + p * 4;
    #pragma unroll
    for (int w = 0; w < 4; ++w) {
        float mw = 0.f;
        #pragma unroll
        for (int j = 0; j < 16; ++j) {
            float tj = 0.f;
            #pragma unroll
            for (int k = 0; k < 16; ++k) tj += Ms[w][j][k] * v[k];
            mw += v[j] * tj;
        }
        fo[w] = mw;
    }
}

// ---------------------------------------------------------------------------
// Kernel 3: logits = feat[2048,784] @ W^T[784,10] + b, then log_softmax.
// One wave (32 lanes) per 16-row tile, N padded to 16, K=784 in 196 steps of
// V_WMMA_F32_16X16X4_F32 (full f32 accumulation). Softmax via 16-lane
// shfl_xor reductions on the D-matrix layout (row = i + 8*half, col = lane%16).
// ---------------------------------------------------------------------------
__global__ void __launch_bounds__(32)
logits_kernel(const float* __restrict__ feat, const float* __restrict__ W,
              const float* __restrict__ bias, float* __restrict__ out) {
    const int lane = threadIdx.x;
    const int half = lane >> 4;   // K sub-pair selector for A/B operands
    const int lm   = lane & 15;   // A: row-in-tile, B/D: column (class)
    const int row0 = blockIdx.x * 16;

    const bool  valid = (lm < 10);
    const float wmask = valid ? 1.f : 0.f;
    const float* arow = feat + (size_t)(row0 + lm) * 784 + 2 * half;
    const float* brow = W + (size_t)(valid ? lm : 0) * 784 + 2 * half; // safe alias, zeroed below

    v8f acc = {0.f, 0.f, 0.f, 0.f, 0.f, 0.f, 0.f, 0.f};
    for (int kk = 0; kk < 784; kk += 4) {
        v2f a = *(const v2f*)(arow + kk);               // A[lane%16][kk + 2*half + {0,1}]
        v2f braw = *(const v2f*)(brow + kk);
        v2f bm; bm[0] = braw[0] * wmask; bm[1] = braw[1] * wmask; // B[kk+2*half+{0,1}][lane%16]
        acc = __builtin_amdgcn_wmma_f32_16x16x4_f32(
            /*neg_a=*/false, a, /*neg_b=*/false, bm,
            /*c_mod=*/(short)0, acc, /*reuse_a=*/false, /*reuse_b=*/false);
    }

    const float bv = valid ? bias[lm] : 0.f;
    union { v8f v; float f[8]; } u; u.v = acc;

    #pragma unroll
    for (int i = 0; i < 8; ++i) {
        const int row = row0 + i + 8 * half;
        const float val = u.f[i] + bv;

        float vmax = valid ? val : -INFINITY;
        vmax = fmaxf(vmax, __shfl_xor(vmax, 1, 32));
        vmax = fmaxf(vmax, __shfl_xor(vmax, 2, 32));
        vmax = fmaxf(vmax, __shfl_xor(vmax, 4, 32));
        vmax = fmaxf(vmax, __shfl_xor(vmax, 8, 32));

        float s = valid ? expf(val - vmax) : 0.f;
        s += __shfl_xor(s, 1, 32);
        s += __shfl_xor(s, 2, 32);
        s += __shfl_xor(s, 4, 32);
        s += __shfl_xor(s, 8, 32);

        if (valid) out[(size_t)row * 10 + lm] = val - vmax - logf(s);
    }
}

// ---------------------------------------------------------------------------
extern "C" void kernel_launch(void* const* d_in, const int* in_sizes, int n_in,
                              void* d_out, int out_size, void* d_ws, size_t ws_size,
                              hipStream_t stream) {
    (void)n_in; (void)out_size; (void)ws_size;
    const float* x    = (const float*)d_in[0]; // [B,784]
    const float* qp   = (const float*)d_in[1]; // [4]
    const float* W    = (const float*)d_in[2]; // [10,784]
    const float* bias = (const float*)d_in[3]; // [10]
    float* out = (float*)d_out;

    float* Mg   = (float*)d_ws;      // 4*16*16 floats
    float* feat = Mg + 1024;         // [B,784] floats

    const int rows  = in_sizes[0] / 784;      // B = 2048
    const int total = rows * NPATCH;          // B*196

    hipLaunchKernelGGL(build_M_kernel, dim3(1), dim3(256), 0, stream, qp, Mg);
    hipLaunchKernelGGL(quanv_kernel, dim3((total + 255) / 256), dim3(256), 0, stream,
                       x, Mg, feat, total);
    hipLaunchKernelGGL(logits_kernel, dim3(rows / 16), dim3(32), 0, stream,
                       feat, W, bias, out);
}